// GAT_1022202216995
// MI455X (gfx1250) — compile-verified
//
#include <hip/hip_runtime.h>
#include <hip/hip_bf16.h>
#include <math.h>

// ---------------------------------------------------------------------------
// Two-layer GAT (PyG GATConv semantics) for MI455X / gfx1250.
// GEMMs use V_WMMA_F32_16X16X4_F32 (fp32-exact matrix core path); each wave
// produces a 16x64 output strip (4 accumulators) so one A-fragment feeds 4
// WMMAs per K-step. Edge softmax/scatter phases are atomic-based.
// ---------------------------------------------------------------------------

#define N_NODES 50000
#define N_EDGES 800000
#define IN_CH   128
#define HEADS   32
#define HID     8
#define HC      (HEADS * HID)   // 256
#define OUT_CH  64
#define NEG_SLOPE 0.2f

typedef float v2f __attribute__((ext_vector_type(2)));
typedef float v8f __attribute__((ext_vector_type(8)));

// ---------------- monotonic float<->uint mapping for atomic max -------------
__device__ __forceinline__ unsigned fflip(float f) {
    unsigned u = __float_as_uint(f);
    return (u & 0x80000000u) ? ~u : (u | 0x80000000u);
}
__device__ __forceinline__ float funflip(unsigned u) {
    return __uint_as_float((u & 0x80000000u) ? (u ^ 0x80000000u) : ~u);
}
__device__ __forceinline__ float lrelu(float x) {
    return x > 0.0f ? x : x * NEG_SLOPE;
}

// ---------------------------------------------------------------------------
// WMMA FP32 GEMM:  C[M,N] = A[M,K] * B[K,N]
// One wave per 16(M) x 64(N) output strip: 4 accumulators share one A-frag
// per K-step of 4 (v_wmma_f32_16x16x4_f32).
// Requires M%16==0, N%64==0, K%4==0 (true for all calls here).
// ---------------------------------------------------------------------------
__global__ void gemm_wmma_f32(const float* __restrict__ A,
                              const float* __restrict__ B,
                              float* __restrict__ C,
                              int M, int N, int K) {
    const int wave = (int)((blockIdx.x * blockDim.x + threadIdx.x) >> 5);
    const int nStrips = N >> 6;                   // strips of 64 columns
    const int mTiles  = M >> 4;
    if (wave >= mTiles * nStrips) return;         // wave-uniform: EXEC stays full
    const int mT = wave / nStrips;
    const int nS = wave % nStrips;

    const int lane = threadIdx.x & 31;
    const int half = lane >> 4;                   // 0: K pair {0,1}, 1: K pair {2,3}
    const int lrow = lane & 15;

    const float* __restrict__ Arow = A + (size_t)(mT * 16 + lrow) * K;
    const float* __restrict__ Bcol = B + (size_t)(nS * 64 + lrow);

    v8f acc0 = {}, acc1 = {}, acc2 = {}, acc3 = {};
#pragma unroll 4
    for (int k0 = 0; k0 < K; k0 += 4) {
        const int ka = k0 + 2 * half;
        v2f a;
        a.x = Arow[ka];
        a.y = Arow[ka + 1];
        const float* __restrict__ b0 = Bcol + (size_t)ka * N;
        const float* __restrict__ b1 = Bcol + (size_t)(ka + 1) * N;
        v2f bf0, bf1, bf2, bf3;
        bf0.x = b0[0];  bf0.y = b1[0];
        bf1.x = b0[16]; bf1.y = b1[16];
        bf2.x = b0[32]; bf2.y = b1[32];
        bf3.x = b0[48]; bf3.y = b1[48];
        acc0 = __builtin_amdgcn_wmma_f32_16x16x4_f32(false, a, false, bf0, (short)0, acc0, false, false);
        acc1 = __builtin_amdgcn_wmma_f32_16x16x4_f32(false, a, false, bf1, (short)0, acc1, false, false);
        acc2 = __builtin_amdgcn_wmma_f32_16x16x4_f32(false, a, false, bf2, (short)0, acc2, false, false);
        acc3 = __builtin_amdgcn_wmma_f32_16x16x4_f32(false, a, false, bf3, (short)0, acc3, false, false);
    }
    // D layout: VGPR r -> row (r + 8*half), col lrow within each 16x16 tile.
    float* __restrict__ Cout = C + (size_t)(mT * 16 + 8 * half) * N + nS * 64 + lrow;
#pragma unroll
    for (int r = 0; r < 8; ++r) {
        float* __restrict__ row = Cout + (size_t)r * N;
        row[0]  = acc0[r];
        row[16] = acc1[r];
        row[32] = acc2[r];
        row[48] = acc3[r];
    }
}

// ---------------------------------------------------------------------------
// Layer-1 per-node attention logits: alpha_src/dst [N,32]
// tid = node*32 + head ; h is [N,32,8] flattened so h + tid*8 is the head slice
// ---------------------------------------------------------------------------
__global__ void alpha1_kernel(const float* __restrict__ h,
                              const float* __restrict__ att_src,
                              const float* __restrict__ att_dst,
                              float* __restrict__ asrc,
                              float* __restrict__ adst) {
    const long long i = (long long)blockIdx.x * blockDim.x + threadIdx.x;
    if (i >= (long long)N_NODES * HEADS) return;
    const int hd = (int)(i & (HEADS - 1));
    const float* __restrict__ hp = h + (size_t)i * HID;
    const float* __restrict__ as = att_src + hd * HID;
    const float* __restrict__ ad = att_dst + hd * HID;
    float s = 0.f, d = 0.f;
#pragma unroll
    for (int c = 0; c < HID; ++c) { s += hp[c] * as[c]; d += hp[c] * ad[c]; }
    asrc[i] = s;
    adst[i] = d;
}

// decode (edge-or-selfloop) -> (s,d) node ids
__device__ __forceinline__ void edge_sd(long long e, const long long* __restrict__ ei,
                                        int& s, int& d) {
    if (e < N_EDGES) {
        s = (int)ei[e];
        d = (int)ei[e + N_EDGES];
    } else {
        s = d = (int)(e - N_EDGES);
    }
}

// ------------------------- layer-1 edge passes ------------------------------
__global__ void edge_max1(const long long* __restrict__ ei,
                          const float* __restrict__ asrc,
                          const float* __restrict__ adst,
                          unsigned* __restrict__ emax) {
    const long long t = (long long)blockIdx.x * blockDim.x + threadIdx.x;
    if (t >= (long long)(N_EDGES + N_NODES) * HEADS) return;
    const int hd = (int)(t & (HEADS - 1));
    int s, d; edge_sd(t >> 5, ei, s, d);
    const float ev = lrelu(asrc[(size_t)s * HEADS + hd] + adst[(size_t)d * HEADS + hd]);
    atomicMax(&emax[(size_t)d * HEADS + hd], fflip(ev));
}

__global__ void edge_sumexp1(const long long* __restrict__ ei,
                             const float* __restrict__ asrc,
                             const float* __restrict__ adst,
                             const unsigned* __restrict__ emax,
                             float* __restrict__ denom) {
    const long long t = (long long)blockIdx.x * blockDim.x + threadIdx.x;
    if (t >= (long long)(N_EDGES + N_NODES) * HEADS) return;
    const int hd = (int)(t & (HEADS - 1));
    int s, d; edge_sd(t >> 5, ei, s, d);
    const size_t di = (size_t)d * HEADS + hd;
    const float ev = lrelu(asrc[(size_t)s * HEADS + hd] + adst[di]);
    atomicAdd(&denom[di], __expf(ev - funflip(emax[di])));
}

__global__ void edge_scatter1(const long long* __restrict__ ei,
                              const float* __restrict__ asrc,
                              const float* __restrict__ adst,
                              const unsigned* __restrict__ emax,
                              const float* __restrict__ denom,
                              const float* __restrict__ h,
                              float* __restrict__ out) {
    const long long t = (long long)blockIdx.x * blockDim.x + threadIdx.x;
    if (t >= (long long)(N_EDGES + N_NODES) * HEADS) return;
    const int hd = (int)(t & (HEADS - 1));
    int s, d; edge_sd(t >> 5, ei, s, d);
    const size_t di = (size_t)d * HEADS + hd;
    const float ev = lrelu(asrc[(size_t)s * HEADS + hd] + adst[di]);
    const float alpha = __expf(ev - funflip(emax[di])) / (denom[di] + 1e-16f);
    const float* __restrict__ hs = h + (size_t)s * HC + hd * HID;
    float* __restrict__ od = out + (size_t)d * HC + hd * HID;
#pragma unroll
    for (int c = 0; c < HID; ++c) atomicAdd(&od[c], hs[c] * alpha);
}

__global__ void bias1_kernel(float* __restrict__ out, const float* __restrict__ b) {
    const long long i = (long long)blockIdx.x * blockDim.x + threadIdx.x;
    if (i >= (long long)N_NODES * HC) return;
    out[i] += b[i & (HC - 1)];
}

// ------------------------- layer-2 (H=1, C=64) ------------------------------
__global__ void alpha2_kernel(const float* __restrict__ h,
                              const float* __restrict__ att_src,
                              const float* __restrict__ att_dst,
                              float* __restrict__ asrc,
                              float* __restrict__ adst) {
    const int n = blockIdx.x * blockDim.x + threadIdx.x;
    if (n >= N_NODES) return;
    const float* __restrict__ hp = h + (size_t)n * OUT_CH;
    float s = 0.f, d = 0.f;
#pragma unroll
    for (int c = 0; c < OUT_CH; ++c) { s += hp[c] * att_src[c]; d += hp[c] * att_dst[c]; }
    asrc[n] = s;
    adst[n] = d;
}

__global__ void edge_max2(const long long* __restrict__ ei,
                          const float* __restrict__ asrc,
                          const float* __restrict__ adst,
                          unsigned* __restrict__ emax) {
    const long long e = (long long)blockIdx.x * blockDim.x + threadIdx.x;
    if (e >= (long long)(N_EDGES + N_NODES)) return;
    int s, d; edge_sd(e, ei, s, d);
    atomicMax(&emax[d], fflip(lrelu(asrc[s] + adst[d])));
}

__global__ void edge_sumexp2(const long long* __restrict__ ei,
                             const float* __restrict__ asrc,
                             const float* __restrict__ adst,
                             const unsigned* __restrict__ emax,
                             float* __restrict__ denom) {
    const long long e = (long long)blockIdx.x * blockDim.x + threadIdx.x;
    if (e >= (long long)(N_EDGES + N_NODES)) return;
    int s, d; edge_sd(e, ei, s, d);
    const float ev = lrelu(asrc[s] + adst[d]);
    atomicAdd(&denom[d], __expf(ev - funflip(emax[d])));
}

// one thread per (edge, group-of-8-channels): recompute scalar alpha, 8 atomics
__global__ void edge_scatter2(const long long* __restrict__ ei,
                              const float* __restrict__ asrc,
                              const float* __restrict__ adst,
                              const unsigned* __restrict__ emax,
                              const float* __restrict__ denom,
                              const float* __restrict__ h,
                              float* __restrict__ out) {
    const long long t = (long long)blockIdx.x * blockDim.x + threadIdx.x;
    if (t >= (long long)(N_EDGES + N_NODES) * 8) return;
    const int g = (int)(t & 7);
    int s, d; edge_sd(t >> 3, ei, s, d);
    const float ev = lrelu(asrc[s] + adst[d]);
    const float alpha = __expf(ev - funflip(emax[d])) / (denom[d] + 1e-16f);
    const float* __restrict__ hs = h + (size_t)s * OUT_CH + g * 8;
    float* __restrict__ od = out + (size_t)d * OUT_CH + g * 8;
#pragma unroll
    for (int c = 0; c < 8; ++c) atomicAdd(&od[c], hs[c] * alpha);
}

__global__ void final_kernel(const float* __restrict__ agg,
                             const float* __restrict__ b,
                             float* __restrict__ out) {
    const long long i = (long long)blockIdx.x * blockDim.x + threadIdx.x;
    if (i >= (long long)N_NODES * OUT_CH) return;
    const float v = agg[i] + b[i & (OUT_CH - 1)];
    out[i] = 1.0f / (1.0f + __expf(-v));
}

// ---------------------------------------------------------------------------
static inline int blocks_for(long long n, int bs) { return (int)((n + bs - 1) / bs); }

extern "C" void kernel_launch(void* const* d_in, const int* in_sizes, int n_in,
                              void* d_out, int out_size, void* d_ws, size_t ws_size,
                              hipStream_t stream) {
    const float*     x        = (const float*)d_in[0];
    const long long* ei       = (const long long*)d_in[1];
    const float*     W1       = (const float*)d_in[2];
    const float*     att_src1 = (const float*)d_in[3];
    const float*     att_dst1 = (const float*)d_in[4];
    const float*     b1       = (const float*)d_in[5];
    const float*     W2       = (const float*)d_in[6];
    const float*     att_src2 = (const float*)d_in[7];
    const float*     att_dst2 = (const float*)d_in[8];
    const float*     b2       = (const float*)d_in[9];
    float*           out      = (float*)d_out;

    // ---- carve workspace ----
    char* p = (char*)d_ws;
    auto carve = [&](size_t bytes) {
        char* r = p;
        p += (bytes + 255) & ~(size_t)255;
        return r;
    };
    float*    h1     = (float*)   carve((size_t)N_NODES * HC * 4);       // [N,256]
    float*    asrc1  = (float*)   carve((size_t)N_NODES * HEADS * 4);    // [N,32]
    float*    adst1  = (float*)   carve((size_t)N_NODES * HEADS * 4);
    unsigned* emax1  = (unsigned*)carve((size_t)N_NODES * HEADS * 4);
    float*    denom1 = (float*)   carve((size_t)N_NODES * HEADS * 4);
    float*    out1   = (float*)   carve((size_t)N_NODES * HC * 4);       // [N,256]
    float*    h2     = (float*)   carve((size_t)N_NODES * OUT_CH * 4);   // [N,64]
    float*    asrc2  = (float*)   carve((size_t)N_NODES * 4);
    float*    adst2  = (float*)   carve((size_t)N_NODES * 4);
    unsigned* emax2  = (unsigned*)carve((size_t)N_NODES * 4);
    float*    denom2 = (float*)   carve((size_t)N_NODES * 4);
    float*    h2agg  = (float*)   carve((size_t)N_NODES * OUT_CH * 4);   // [N,64]

    const int BS = 256;

    // zero accumulators (fflip of any finite > 0 -> memset(0) is a safe -inf)
    hipMemsetAsync(emax1,  0, (size_t)N_NODES * HEADS * 4, stream);
    hipMemsetAsync(denom1, 0, (size_t)N_NODES * HEADS * 4, stream);
    hipMemsetAsync(out1,   0, (size_t)N_NODES * HC * 4,    stream);
    hipMemsetAsync(emax2,  0, (size_t)N_NODES * 4,         stream);
    hipMemsetAsync(denom2, 0, (size_t)N_NODES * 4,         stream);
    hipMemsetAsync(h2agg,  0, (size_t)N_NODES * OUT_CH * 4, stream);

    // ---- layer 1 ----
    {   // h1 = x @ W1   (M=50000, N=256, K=128) ; 16x64 strip per wave
        const long long waves = (long long)(N_NODES / 16) * (HC / 64);
        gemm_wmma_f32<<<blocks_for(waves * 32, BS), BS, 0, stream>>>(
            x, W1, h1, N_NODES, HC, IN_CH);
    }
    alpha1_kernel<<<blocks_for((long long)N_NODES * HEADS, BS), BS, 0, stream>>>(
        h1, att_src1, att_dst1, asrc1, adst1);

    const long long eh = (long long)(N_EDGES + N_NODES) * HEADS;
    edge_max1   <<<blocks_for(eh, BS), BS, 0, stream>>>(ei, asrc1, adst1, emax1);
    edge_sumexp1<<<blocks_for(eh, BS), BS, 0, stream>>>(ei, asrc1, adst1, emax1, denom1);
    edge_scatter1<<<blocks_for(eh, BS), BS, 0, stream>>>(ei, asrc1, adst1, emax1, denom1,
                                                         h1, out1);
    bias1_kernel<<<blocks_for((long long)N_NODES * HC, BS), BS, 0, stream>>>(out1, b1);

    // ---- layer 2 ----
    {   // h2 = out1 @ W2  (M=50000, N=64, K=256) ; one strip covers all 64 cols
        const long long waves = (long long)(N_NODES / 16) * (OUT_CH / 64);
        gemm_wmma_f32<<<blocks_for(waves * 32, BS), BS, 0, stream>>>(
            out1, W2, h2, N_NODES, OUT_CH, HC);
    }
    alpha2_kernel<<<blocks_for(N_NODES, BS), BS, 0, stream>>>(
        h2, att_src2, att_dst2, asrc2, adst2);

    const long long e1 = (long long)(N_EDGES + N_NODES);
    edge_max2   <<<blocks_for(e1, BS), BS, 0, stream>>>(ei, asrc2, adst2, emax2);
    edge_sumexp2<<<blocks_for(e1, BS), BS, 0, stream>>>(ei, asrc2, adst2, emax2, denom2);
    edge_scatter2<<<blocks_for(e1 * 8, BS), BS, 0, stream>>>(ei, asrc2, adst2, emax2,
                                                             denom2, h2, h2agg);

    final_kernel<<<blocks_for((long long)N_NODES * OUT_CH, BS), BS, 0, stream>>>(
        h2agg, b2, out);
}